// OuterProduct_2319282340355
// MI455X (gfx1250) — compile-verified
//
#include <hip/hip_runtime.h>

typedef __attribute__((ext_vector_type(2))) float v2f;
typedef __attribute__((ext_vector_type(4))) float v4f;
typedef __attribute__((ext_vector_type(8))) float v8f;

// Per batch: s = sum over 64 rows of x[b] (64x128), out[b] = s s^T (128x128).
// One 256-thread block (8 wave32) per batch.
//   Phase 1: coalesced float4 reduction of x[b] -> s[128] in LDS.
//   Phase 2: wave w computes output tile-row w; each 16x16 tile is one
//            V_WMMA_F32_16X16X4_F32 rank-1 update (A has s_i in K=0 slot,
//            B has s_j in K=0 row, all other K slots zero).
__global__ __launch_bounds__(256) void outer_product_wmma_kernel(
    const float* __restrict__ x, float* __restrict__ out) {
  const int b    = blockIdx.x;
  const int t    = threadIdx.x;       // 0..255
  const int lane = t & 31;
  const int wave = t >> 5;            // 0..7

  __shared__ float partial[8][128];   // 4 KB: per-rowblock partial sums
  __shared__ float s[128];            // final column sums

  const float* __restrict__ xb = x + (size_t)b * (64 * 128);

  // ---- Phase 1: reduction over the 64-row axis ----
  // Thread t handles float4 column group cg = t&31 (cols 4*cg..4*cg+3)
  // and row block rb = t>>5 (rows rb*8 .. rb*8+7).
  {
    const int cg = t & 31;
    const int rb = t >> 5;
    v4f acc = {0.f, 0.f, 0.f, 0.f};
#pragma unroll
    for (int r = 0; r < 8; ++r) {
      const v4f v = *(const v4f*)(xb + (size_t)(rb * 8 + r) * 128 + cg * 4);
      acc += v;
    }
    partial[rb][cg * 4 + 0] = acc.x;
    partial[rb][cg * 4 + 1] = acc.y;
    partial[rb][cg * 4 + 2] = acc.z;
    partial[rb][cg * 4 + 3] = acc.w;
  }
  __syncthreads();
  if (t < 128) {
    float v = 0.f;
#pragma unroll
    for (int k = 0; k < 8; ++k) v += partial[k][t];
    s[t] = v;
  }
  __syncthreads();

  // ---- Phase 2: outer product via WMMA f32 16x16x4 (rank-1, K=0 only) ----
  // A-matrix 16x4 f32 layout: VGPR0 lanes 0-15 = (M=lane, K=0). Zero K=1..3.
  // B-matrix 4x16 f32 layout: VGPR0 lanes 0-15 = (K=0, N=lane). Zero rest.
  // C/D 16x16 f32 layout: VGPR r -> M=r (lanes 0-15), M=r+8 (lanes 16-31).
  const int  n16   = lane & 15;
  const bool lowHi = (lane < 16);

  v2f A;
  A.x = lowHi ? s[wave * 16 + n16] : 0.0f;   // s_i slice for this wave's tile row
  A.y = 0.0f;

  float* __restrict__ outb = out + (size_t)b * (128 * 128);
  const int mbase = wave * 16 + ((lane >> 4) << 3);   // +8 for upper half lanes

#pragma unroll
  for (int j = 0; j < 8; ++j) {
    v2f Bv;
    Bv.x = lowHi ? s[j * 16 + n16] : 0.0f;   // s_j slice for tile column j
    Bv.y = 0.0f;

    v8f c = {};
    // 8 args: (neg_a, A, neg_b, B, c_mod, C, reuse_a, reuse_b)
    c = __builtin_amdgcn_wmma_f32_16x16x4_f32(
        /*neg_a=*/false, A, /*neg_b=*/false, Bv,
        /*c_mod=*/(short)0, c, /*reuse_a=*/false, /*reuse_b=*/false);

#pragma unroll
    for (int r = 0; r < 8; ++r) {
      // lane L, VGPR r holds element (M = mbase + r, N = j*16 + n16)
      outb[(size_t)(mbase + r) * 128 + j * 16 + n16] = c[r];
    }
  }
}

extern "C" void kernel_launch(void* const* d_in, const int* in_sizes, int n_in,
                              void* d_out, int out_size, void* d_ws, size_t ws_size,
                              hipStream_t stream) {
  (void)n_in; (void)out_size; (void)d_ws; (void)ws_size;
  const float* x = (const float*)d_in[0];
  float* out = (float*)d_out;
  const int B = in_sizes[0] / (64 * 128);   // 4096 for the reference shapes
  outer_product_wmma_kernel<<<dim3(B), dim3(256), 0, stream>>>(x, out);
}